// NMSFilter_86122684219468
// MI455X (gfx1250) — compile-verified
//
#include <hip/hip_runtime.h>
#include <hip/hip_bf16.h>
#include <stdint.h>

#define NN   2048
#define BB   8
#define CC   32
#define WPR  64            // 32-bit words per mask row (2048 bits)
#define NMS_T 0.45f
#define PRE_T 0.005f

typedef float v2f __attribute__((ext_vector_type(2)));
typedef float v8f __attribute__((ext_vector_type(8)));
typedef unsigned int v4u __attribute__((ext_vector_type(4)));
typedef int v4i __attribute__((ext_vector_type(4)));
typedef int v8i __attribute__((ext_vector_type(8)));

#if defined(__AMDGCN__)
// Select one of 8 wave-uniform scalars by (k & 7) with a cndmask tree
// (avoids scratch spill from dynamic indexing).
__device__ __forceinline__ unsigned sel8(unsigned b0, unsigned b1, unsigned b2, unsigned b3,
                                         unsigned b4, unsigned b5, unsigned b6, unsigned b7,
                                         int k) {
    unsigned x01 = (k & 1) ? b1 : b0;
    unsigned x23 = (k & 1) ? b3 : b2;
    unsigned x45 = (k & 1) ? b5 : b4;
    unsigned x67 = (k & 1) ? b7 : b6;
    unsigned y0  = (k & 2) ? x23 : x01;
    unsigned y1  = (k & 2) ? x67 : x45;
    return (k & 4) ? y1 : y0;
}
#endif

// -------------------------------------------------------------------------
// Kernel A: build per-image suppression bitmask  mask[b][i][w] (w in [0,64))
//   bit j of row i  =  (IoU(box_i, box_j) > 0.45)
// One wave owns 16 rows; sweeps 128 column tiles of 16.
// Union's (area_i + area_j) term computed on the matrix pipe via
// V_WMMA_F32_16X16X4_F32 as an outer-product-sum; intersection on VALU.
// Box table (32 KB) staged into LDS by the Tensor Data Mover.
// -------------------------------------------------------------------------
__global__ __launch_bounds__(256) void nms_mask_kernel(const float* __restrict__ bbs,
                                                       unsigned* __restrict__ mask) {
#if defined(__AMDGCN__)
    __shared__ float4 sbox[NN];           // 32 KB
    const int b    = blockIdx.x >> 4;
    const int tid  = threadIdx.x;
    const int lane = tid & 31;
    const int wave = tid >> 5;
    const float* bbs_b = bbs + (size_t)b * NN * 4;

#if defined(__gfx1250__) && __has_builtin(__builtin_amdgcn_tensor_load_to_lds)
    if (tid < 32) {   // wave 0 issues one TDM descriptor: 8192 f32 (1-row tile)
        unsigned lds_off = (unsigned)(uintptr_t)&sbox[0];          // flat low 32 = LDS offset
        unsigned long long ga = (unsigned long long)(uintptr_t)bbs_b;
        v4u g0;
        g0.x = 0x1u;                                               // count=1, user descriptor
        g0.y = lds_off;                                            // lds_addr
        g0.z = (unsigned)ga;                                       // global_addr[31:0]
        g0.w = ((unsigned)(ga >> 32) & 0x1FFFFFFu) | (2u << 30);   // addr[56:32] | type=2
        v8i g1;
        g1[0] = (int)(2u << 16);          // data_size = 4B
        g1[1] = (int)(0x2000u << 16);     // tensor_dim0 = 8192 (bits 63:48)
        g1[2] = (int)(1u << 16);          // dim0 hi16=0 ; tensor_dim1 = 1 (bits 95:80)
        g1[3] = (int)(0x2000u << 16);     // dim1 hi16=0 ; tile_dim0 = 8192 (bits 127:112)
        g1[4] = 1;                        // tile_dim1 = 1 ; tile_dim2 = 0
        g1[5] = 0x2000;                   // tensor_dim0_stride = 8192 (bits 191:160)
        g1[6] = (int)(0x2000u << 16);     // stride0 hi16=0 ; tensor_dim1_stride lo16
        g1[7] = 0;                        // tensor_dim1_stride hi32 = 0
        v4i z4 = {0, 0, 0, 0};
#if __clang_major__ >= 23
        v8i z8 = {0, 0, 0, 0, 0, 0, 0, 0};
        __builtin_amdgcn_tensor_load_to_lds(g0, g1, z4, z4, z8, 0);
#else
        __builtin_amdgcn_tensor_load_to_lds(g0, g1, z4, z4, 0);
#endif
        __builtin_amdgcn_s_wait_tensorcnt(0);
    }
#else
    {
        const float4* src = (const float4*)bbs_b;
        for (int i = tid; i < NN; i += 256) sbox[i] = src[i];
    }
#endif
    __syncthreads();

    const int i0 = (((blockIdx.x & 15) * 8) + wave) * 16;   // 16-row stripe base
    const int hi = lane >> 4;

    // 8 row boxes per lane (rows i0 + 8*hi + v), fixed for the whole sweep.
    float4 rb[8];
    float  rowArea[8];
#pragma unroll
    for (int v = 0; v < 8; ++v) {
        rb[v] = sbox[i0 + hi * 8 + v];
        rowArea[v] = (rb[v].z - rb[v].x) * (rb[v].w - rb[v].y);
    }

    // WMMA A operand: lanes 0..15 hold (area_row[M=lane], 1); lanes 16..31 zero (K=2,3).
    float4 abx = sbox[i0 + (lane & 15)];
    float  areaA = (abx.z - abx.x) * (abx.w - abx.y);
    v2f aop;
    aop.x = (lane < 16) ? areaA : 0.0f;
    aop.y = (lane < 16) ? 1.0f  : 0.0f;

    unsigned* mrow = mask + (size_t)(b * NN + i0) * WPR;

    for (int jt = 0; jt < 128; jt += 2) {
        unsigned bal[2][8];
#pragma unroll
        for (int s = 0; s < 2; ++s) {
            const int j = (jt + s) * 16 + (lane & 15);
            float4 cb = sbox[j];
            float areaB = (cb.z - cb.x) * (cb.w - cb.y);

            v8f acc = {0, 0, 0, 0, 0, 0, 0, 0};
#if defined(__gfx1250__) && __has_builtin(__builtin_amdgcn_wmma_f32_16x16x4_f32)
            // D[M][N] = area_i(M) * 1 + 1 * area_j(N)  (exact fp32, RNE == a+b)
            v2f bop;
            bop.x = (lane < 16) ? 1.0f  : 0.0f;
            bop.y = (lane < 16) ? areaB : 0.0f;
            acc = __builtin_amdgcn_wmma_f32_16x16x4_f32(false, aop, false, bop,
                                                        (short)0, acc, false, false);
#else
#pragma unroll
            for (int v = 0; v < 8; ++v) acc[v] = rowArea[v] + areaB;
#endif
#pragma unroll
            for (int v = 0; v < 8; ++v) {
                float ltx = fmaxf(rb[v].x, cb.x);
                float lty = fmaxf(rb[v].y, cb.y);
                float rbx = fminf(rb[v].z, cb.z);
                float rby = fminf(rb[v].w, cb.w);
                float w  = fmaxf(rbx - ltx, 0.0f);
                float h  = fmaxf(rby - lty, 0.0f);
                float inter = w * h;
                float uni   = acc[v] - inter;                 // area_i + area_j - inter
                float q     = inter / fmaxf(uni, 1e-12f);     // exact reference arithmetic
                bal[s][v]   = __builtin_amdgcn_ballot_w32(q > NMS_T);
            }
        }
        // Pack: ballot(v) = { rows v (bits 0..15) , v+8 (bits 16..31) } over 16 cols.
        // Lanes 0..15 each store one 32-bit word for row r=lane at word jt/2.
        int r = lane & 15;
        unsigned be = sel8(bal[0][0], bal[0][1], bal[0][2], bal[0][3],
                           bal[0][4], bal[0][5], bal[0][6], bal[0][7], r & 7);
        unsigned bo = sel8(bal[1][0], bal[1][1], bal[1][2], bal[1][3],
                           bal[1][4], bal[1][5], bal[1][6], bal[1][7], r & 7);
        unsigned wrd = (r < 8) ? ((be & 0xFFFFu) | (bo << 16))
                               : ((be >> 16) | (bo & 0xFFFF0000u));
        if (lane < 16) mrow[(size_t)r * WPR + (jt >> 1)] = wrd;
    }
#endif // __AMDGCN__
}

// -------------------------------------------------------------------------
// Kernel B: one workgroup per (b,c).
//   1) stable descending bitonic sort of (conf, idx) packed in u64 keys (LDS)
//   2) wave 0 runs the greedy chain: suppression bitset = 64 words (2/lane),
//      active step ORs one 256-B mask row from L2; rows prefetched ahead.
// -------------------------------------------------------------------------
__global__ __launch_bounds__(256) void nms_seq_kernel(const float* __restrict__ conf,
                                                      const unsigned* __restrict__ mask,
                                                      float* __restrict__ out) {
#if defined(__AMDGCN__)
    __shared__ unsigned long long keys[NN];   // 16 KB
    __shared__ unsigned sup[WPR];
    const int bc  = blockIdx.x;               // b*32 + c
    const int b   = bc >> 5;
    const int tid = threadIdx.x;
    const float* cin  = conf + (size_t)bc * NN;
    float*       cout = out  + (size_t)bc * NN;

    for (int n = tid; n < NN; n += 256) {
        float c = cin[n];
        c = (c > PRE_T) ? c : 0.0f;           // pre-threshold (conf >= 0 so bits monotonic)
        keys[n] = ((unsigned long long)__float_as_uint(c) << 32) |
                  (unsigned)(NN - 1 - n);     // stable: ties -> smaller index first
    }
    __syncthreads();

    // Bitonic sort, descending.
    for (unsigned k = 2; k <= NN; k <<= 1) {
        for (unsigned j = k >> 1; j > 0; j >>= 1) {
            for (int e = 0; e < NN / 256; ++e) {
                unsigned i   = (unsigned)tid + (unsigned)e * 256u;
                unsigned ixj = i ^ j;
                if (ixj > i) {
                    unsigned long long a  = keys[i];
                    unsigned long long bk = keys[ixj];
                    bool up = ((i & k) == 0);
                    if (up ? (a < bk) : (a > bk)) { keys[i] = bk; keys[ixj] = a; }
                }
            }
            __syncthreads();
        }
    }

    if (tid < 32) {
        const int lane = tid;
        sup[2 * lane]     = 0;
        sup[2 * lane + 1] = 0;
        const unsigned* mb = mask + (size_t)b * NN * WPR;
        unsigned s0 = 0, s1 = 0;
        for (int i = 0; i < NN; ++i) {
            unsigned long long key = keys[i];
            float cs = __uint_as_float((unsigned)(key >> 32));
            int   oi = NN - 1 - (int)(unsigned)(key & 0xFFFFFFFFull);
            if (i + 8 < NN) {      // hide L2 latency on the dependent chain
                int on = NN - 1 - (int)(unsigned)(keys[i + 8] & 0xFFFFFFFFull);
                __builtin_prefetch((const void*)(mb + (size_t)on * WPR + 2 * lane), 0, 1);
            }
            unsigned sb = (sup[oi >> 5] >> (oi & 31)) & 1u;
            bool active = (sb == 0u) && (cs > 0.0f);
            if (active) {
                const unsigned* row = mb + (size_t)oi * WPR;
                s0 |= row[2 * lane];
                s1 |= row[2 * lane + 1];
                sup[2 * lane]     = s0;
                sup[2 * lane + 1] = s1;
            }
            if (lane == 0) cout[oi] = active ? cs : 0.0f;
        }
    }
#endif // __AMDGCN__
}

extern "C" void kernel_launch(void* const* d_in, const int* in_sizes, int n_in,
                              void* d_out, int out_size, void* d_ws, size_t ws_size,
                              hipStream_t stream) {
    const float* bbs  = (const float*)d_in[0];   // [B,N,4]
    const float* conf = (const float*)d_in[1];   // [B,C,N]
    float* out = (float*)d_out;                  // [B,C,N]
    unsigned* mask = (unsigned*)d_ws;            // B*N*64 u32 = 4 MB (L2-resident)

    nms_mask_kernel<<<dim3(BB * 16), dim3(256), 0, stream>>>(bbs, mask);
    nms_seq_kernel<<<dim3(BB * CC), dim3(256), 0, stream>>>(conf, mask, out);
}